// SPGG_Fermi_7902739824972
// MI455X (gfx1250) — compile-verified
//
#include <hip/hip_runtime.h>
#include <stdint.h>

// SPGG Fermi update, fused single pass, L=4096, R=4.9, K=0.1.
//
// Bit-fidelity notes vs the JAX reference:
//  * type in {0,1} exactly -> c==t, d==1-t; profit = c5*c + d5*d == select(t, c5, d5)
//    == sum_plus(dp_i - t) with the reference's left association (x-0.0f == x).
//  * coop_num in {0..5} exactly -> coop/5.0f == coop*0.2f bit-exactly for those values.
//  * plus_sum order mirrored: self + up + down + left + right, left-assoc.

#define L_DIM   4096
#define L_MASK  4095
#define TILE    64
#define NTHREADS 256

// type tile: rows halo 3 (70 rows), cols halo 4 left / 4 right (72 cols) so that
// both global and LDS addresses of each row chunk are 16B aligned for b128 DMA.
#define T_ROWS  70
#define T_COLS  72
#define T_PITCH 72            // floats; 288B rows -> 16B-aligned b128 stores
#define OFFX    4             // lds col 0 == global col (base_c - 4)
#define OFFY    3             // lds row 0 == global row (base_r - 3)

#define C_W     68            // coop_num, halo 2
#define C_PITCH 69
#define P_W     66            // profit, halo 1
#define P_PITCH 67

__launch_bounds__(NTHREADS)
__global__ void spgg_fermi_kernel(const float* __restrict__ type_m,
                                  const int*   __restrict__ ldir,
                                  const float* __restrict__ lprob,
                                  float* __restrict__ out_type,
                                  float* __restrict__ out_profit)
{
    __shared__ float lds_t[T_ROWS * T_PITCH];  // type
    __shared__ float lds_c[C_W * C_PITCH];     // coop_num
    __shared__ float lds_p[P_W * P_PITCH];     // profit

    const int tid    = threadIdx.x;
    const int base_r = blockIdx.y * TILE;
    const int base_c = blockIdx.x * TILE;

    // ---- Stage 1: async DMA type tile into LDS (ASYNCcnt-tracked, no VGPR data) ----
    if (blockIdx.x != 0 && blockIdx.x != (L_DIM / TILE - 1)) {
        // Fast path: no column wrap. 18 x b128 per row, 16B aligned both sides.
        for (int i = tid; i < T_ROWS * 18; i += NTHREADS) {
            int ty  = i / 18;
            int c16 = i - ty * 18;                    // 16B chunk within row
            int gr  = (base_r + ty - OFFY) & L_MASK;  // periodic row wrap
            uint64_t gaddr = (uint64_t)(uintptr_t)
                (type_m + (((size_t)gr << 12) + (size_t)(base_c - OFFX) + ((size_t)c16 << 2)));
            uint32_t laddr = (uint32_t)(uintptr_t)(&lds_t[ty * T_PITCH + (c16 << 2)]);
            asm volatile("global_load_async_to_lds_b128 %0, %1, off"
                         :: "v"(laddr), "v"(gaddr) : "memory");
        }
    } else {
        // Edge blocks: per-element b32 with full periodic wrap in both dims.
        for (int i = tid; i < T_ROWS * T_COLS; i += NTHREADS) {
            int ty = i / T_COLS;
            int tx = i - ty * T_COLS;
            int gr = (base_r + ty - OFFY) & L_MASK;
            int gc = (base_c + tx - OFFX) & L_MASK;
            uint64_t gaddr = (uint64_t)(uintptr_t)(type_m + (((size_t)gr << 12) + gc));
            uint32_t laddr = (uint32_t)(uintptr_t)(&lds_t[ty * T_PITCH + tx]);
            asm volatile("global_load_async_to_lds_b32 %0, %1, off"
                         :: "v"(laddr), "v"(gaddr) : "memory");
        }
    }
    asm volatile("s_wait_asynccnt 0" ::: "memory");
    __syncthreads();

    // ---- Stage 2: coop_num = plus_sum(t) on tile + halo2 (c == t exactly) ----
    // coop(cy,cx) <-> global (base_r + cy - 2, base_c + cx - 2) -> lds_t (cy+1, cx+2)
    for (int i = tid; i < C_W * C_W; i += NTHREADS) {
        int cy = i / C_W;
        int cx = i - cy * C_W;
        const float* t0 = &lds_t[(cy + 1) * T_PITCH + (cx + 2)];
        lds_c[cy * C_PITCH + cx] =
            (((t0[0] + t0[-T_PITCH]) + t0[T_PITCH]) + t0[-1]) + t0[1];
    }
    __syncthreads();

    // ---- Stage 3: profit on tile + halo1 ----
    // dp = coop*0.2f*4.9f (== coop/5.0f*4.9f for coop in 0..5); profit = sum(dp_i - t)
    for (int i = tid; i < P_W * P_W; i += NTHREADS) {
        int py = i / P_W;
        int px = i - py * P_W;
        const float* c0 = &lds_c[(py + 1) * C_PITCH + (px + 1)];
        float t   = lds_t[(py + 2) * T_PITCH + (px + 3)];   // 0.0f or 1.0f
        float dps = (c0[0]        * 0.2f) * 4.9f;
        float dpu = (c0[-C_PITCH] * 0.2f) * 4.9f;
        float dpd = (c0[ C_PITCH] * 0.2f) * 4.9f;
        float dpl = (c0[-1]       * 0.2f) * 4.9f;
        float dpr = (c0[ 1]       * 0.2f) * 4.9f;
        lds_p[py * P_PITCH + px] =
            ((((dps - t) + (dpu - t)) + (dpd - t)) + (dpl - t)) + (dpr - t);
    }
    __syncthreads();

    // ---- Stage 4: Fermi imitation on the 64x64 core, write both planes ----
    for (int i = tid; i < TILE * TILE; i += NTHREADS) {
        int oy = i >> 6;
        int ox = i & 63;
        size_t g = (size_t)(base_r + oy) * L_DIM + (base_c + ox);

        float p   = lds_p[(oy + 1) * P_PITCH + (ox + 1)];
        int   dir = ldir[g];
        // 0=left(j-1), 1=right(j+1), 2=up(i-1), 3=down(i+1)
        int dro = (dir == 2) ? -1 : ((dir == 3) ? 1 : 0);
        int dco = (dir == 0) ? -1 : ((dir == 1) ? 1 : 0);
        float pn = lds_p[(oy + 1 + dro) * P_PITCH + (ox + 1 + dco)];
        float tn = lds_t[(oy + OFFY + dro) * T_PITCH + (ox + OFFX + dco)];
        float ts = lds_t[(oy + OFFY) * T_PITCH + (ox + OFFX)];

        // W = sigmoid((pn - p)/K), K = 0.1
        float w  = 1.0f / (1.0f + __expf((p - pn) * 10.0f));
        float lp = lprob[g];

        out_type[g]   = (lp <= w) ? tn : ts;
        out_profit[g] = p;
    }
}

extern "C" void kernel_launch(void* const* d_in, const int* in_sizes, int n_in,
                              void* d_out, int out_size, void* d_ws, size_t ws_size,
                              hipStream_t stream) {
    const float* type_m = (const float*)d_in[0];
    const int*   ldir   = (const int*)d_in[1];
    const float* lprob  = (const float*)d_in[2];
    float* out = (float*)d_out;

    dim3 grid(L_DIM / TILE, L_DIM / TILE);
    spgg_fermi_kernel<<<grid, dim3(NTHREADS), 0, stream>>>(
        type_m, ldir, lprob,
        /*plane 0: type_t1 */ out,
        /*plane 1: profit  */ out + (size_t)L_DIM * L_DIM);
}